// NeighborConstrainedRegLoss_70033736728623
// MI455X (gfx1250) — compile-verified
//
#include <hip/hip_runtime.h>
#include <hip/hip_bf16.h>

typedef float v2f __attribute__((ext_vector_type(2)));
typedef float v8f __attribute__((ext_vector_type(8)));

#define N_CLASSES 16

// ---------------------------------------------------------------------------
// Zero workspace: sums[N*C], cnt[N], accum[2] (double: term1, term2)
// ---------------------------------------------------------------------------
__global__ void ncr_zero_kernel(float* __restrict__ sums, float* __restrict__ cnt,
                                double* __restrict__ accum, int n_sums, int n_cnt) {
  int stride = gridDim.x * blockDim.x;
  int tid = blockIdx.x * blockDim.x + threadIdx.x;
  for (int k = tid; k < n_sums; k += stride) sums[k] = 0.0f;
  for (int k = tid; k < n_cnt; k += stride) cnt[k] = 0.0f;
  if (tid < 2) accum[tid] = 0.0;
}

// ---------------------------------------------------------------------------
// Scatter-mean numerators + degree counts. One thread per edge; node rows are
// L2-resident (6.4 MB << 192 MB) so the 17 atomics/edge hit L2.
// ---------------------------------------------------------------------------
__global__ void ncr_scatter_kernel(const int* __restrict__ src, const int* __restrict__ dst,
                                   const float* __restrict__ pred,
                                   float* __restrict__ sums, float* __restrict__ cnt, int E) {
  int e = blockIdx.x * blockDim.x + threadIdx.x;
  if (e >= E) return;
  int s = src[e];
  int d = dst[e];
  const float4* p = reinterpret_cast<const float4*>(pred + (size_t)s * N_CLASSES);
  float* o = sums + (size_t)d * N_CLASSES;
  float4 v;
  v = p[0]; atomicAdd(o + 0, v.x);  atomicAdd(o + 1, v.y);  atomicAdd(o + 2, v.z);  atomicAdd(o + 3, v.w);
  v = p[1]; atomicAdd(o + 4, v.x);  atomicAdd(o + 5, v.y);  atomicAdd(o + 6, v.z);  atomicAdd(o + 7, v.w);
  v = p[2]; atomicAdd(o + 8, v.x);  atomicAdd(o + 9, v.y);  atomicAdd(o + 10, v.z); atomicAdd(o + 11, v.w);
  v = p[3]; atomicAdd(o + 12, v.x); atomicAdd(o + 13, v.y); atomicAdd(o + 14, v.z); atomicAdd(o + 15, v.w);
  atomicAdd(cnt + d, 1.0f);
}

// ---------------------------------------------------------------------------
// Per-node transform. Reads sums (numerators) + cnt, produces in-place:
//   buf[n,c]  = log(avg + 1e-10)              (LS, gathered by src later)
//   Dmat[n,c] = sharp + 1e-10                 (D, gathered by dst later)
// and folds Σ_e Σ_c d·log d = Σ_n indeg(n)·Σ_c d·log d into accum[0].
// ---------------------------------------------------------------------------
__global__ void ncr_node_kernel(float* __restrict__ buf, const float* __restrict__ cnt,
                                float* __restrict__ Dmat, double* __restrict__ accum, int N) {
  int n = blockIdx.x * blockDim.x + threadIdx.x;
  if (n >= N) return;
  float c_raw = cnt[n];
  float inv = 1.0f / fmaxf(c_raw, 1.0f);
  float* row = buf + (size_t)n * N_CLASSES;
  float* drow = Dmat + (size_t)n * N_CLASSES;
  float avg[N_CLASSES], p[N_CLASSES];
  float psum = 0.0f;
#pragma unroll
  for (int k = 0; k < N_CLASSES; ++k) {
    avg[k] = row[k] * inv;         // scatter-mean
    p[k] = avg[k] * avg[k];        // avg ** (1/TEMP), TEMP=0.5
    psum += p[k];
  }
  float rinv = 1.0f / psum;
  float h = 0.0f;
#pragma unroll
  for (int k = 0; k < N_CLASSES; ++k) {
    float d = p[k] * rinv + 1e-10f;   // sharp_avg_pred + eps
    h += d * __logf(d);
    drow[k] = d;
    row[k] = __logf(avg[k] + 1e-10f); // log(avg_pred + eps)
  }
  atomicAdd(accum + 0, (double)(c_raw * h));
}

// ---------------------------------------------------------------------------
// Edge cross-term via f32 WMMA: per wave, a tile of 16 edges.
//   A (16x4 chunk)  = D[dst[e]] rows,  B (4x16 chunk) = LS[src[e]] cols.
// After 4 K-chunks, diag(acc) = dot(D[dst[e]], LS[src[e]]) for 16 edges
// in full f32 precision. Diagonal of the 16x16 f32 C/D layout:
//   acc[r] @ lane r  (M=N=r)  and  acc[r] @ lane r+24 (M=N=r+8).
// ---------------------------------------------------------------------------
__global__ void ncr_edge_wmma_kernel(const int* __restrict__ src, const int* __restrict__ dst,
                                     const float* __restrict__ Dmat, const float* __restrict__ LS,
                                     double* __restrict__ accum, int E) {
  const int lane = threadIdx.x & 31;
  const int wave = (blockIdx.x * blockDim.x + threadIdx.x) >> 5;
  const int n_waves = (gridDim.x * blockDim.x) >> 5;
  const int n_tiles = (E + 15) >> 4;

  const int m = lane & 15;    // A-row index == B-col index for this lane
  const int hi = lane >> 4;   // K half select per ISA f32 A/B layout

  for (int t = wave; t < n_tiles; t += n_waves) {
    int e = t * 16 + m;
    float w = (e < E) ? 1.0f : 0.0f;   // zero-weight tail, keep EXEC full
    int ec = (e < E) ? e : (E - 1);
    const float* drow = Dmat + (size_t)dst[ec] * N_CLASSES;  // A source
    const float* lrow = LS + (size_t)src[ec] * N_CLASSES;    // B source

    v8f acc = {};
#pragma unroll
    for (int j = 0; j < 4; ++j) {
      int k0 = 4 * j + (hi ? 2 : 0);
      v2f a, b;
      a.x = drow[k0] * w;
      a.y = drow[k0 + 1] * w;
      b.x = lrow[k0];
      b.y = lrow[k0 + 1];
      acc = __builtin_amdgcn_wmma_f32_16x16x4_f32(
          /*neg_a=*/false, a, /*neg_b=*/false, b,
          /*c_mod=*/(short)0, acc, /*reuse_a=*/false, /*reuse_b=*/false);
    }

    // Pull the 16 diagonal entries (per-edge losses) and wave-reduce.
    float contrib = 0.0f;
#pragma unroll
    for (int r = 0; r < 8; ++r) {
      if (lane == r || lane == r + 24) contrib += acc[r];
    }
#pragma unroll
    for (int off = 16; off >= 1; off >>= 1)
      contrib += __shfl_xor(contrib, off, 32);
    if (lane == 0) atomicAdd(accum + 1, (double)contrib);
  }
}

// ---------------------------------------------------------------------------
// out = (term1 - term2) / E
// ---------------------------------------------------------------------------
__global__ void ncr_finalize_kernel(const double* __restrict__ accum, float* __restrict__ out, int E) {
  if (blockIdx.x == 0 && threadIdx.x == 0)
    out[0] = (float)((accum[0] - accum[1]) / (double)E);
}

extern "C" void kernel_launch(void* const* d_in, const int* in_sizes, int n_in,
                              void* d_out, int out_size, void* d_ws, size_t ws_size,
                              hipStream_t stream) {
  const int* edge = (const int*)d_in[0];       // [2, E] int32
  const float* pred = (const float*)d_in[1];   // [N, 16] float32
  const int E = in_sizes[0] / 2;
  const int N = in_sizes[1] / N_CLASSES;
  const int* src = edge;
  const int* dst = edge + E;

  // Workspace layout (256B-aligned blocks):
  //   [0)        accum : 2 doubles (term1, term2)
  //   [256)      buf   : N*16 f32   sums -> log_s (in place)
  //   [+N*64)    Dmat  : N*16 f32
  //   [+2*N*64)  cnt   : N f32
  char* ws = (char*)d_ws;
  double* accum = (double*)ws;
  float* buf = (float*)(ws + 256);
  float* Dmat = (float*)(ws + 256 + (size_t)N * N_CLASSES * sizeof(float));
  float* cnt = (float*)(ws + 256 + 2 * (size_t)N * N_CLASSES * sizeof(float));

  const int n_sums = N * N_CLASSES;
  int zb = (n_sums + 255) / 256;
  if (zb > 2048) zb = 2048;
  ncr_zero_kernel<<<zb, 256, 0, stream>>>(buf, cnt, accum, n_sums, N);

  ncr_scatter_kernel<<<(E + 127) / 128, 128, 0, stream>>>(src, dst, pred, buf, cnt, E);

  ncr_node_kernel<<<(N + 127) / 128, 128, 0, stream>>>(buf, cnt, Dmat, accum, N);

  const int n_tiles = (E + 15) / 16;
  const int waves_per_block = 8;  // 256 threads = 8 wave32
  int blocks = (n_tiles + waves_per_block - 1) / waves_per_block;
  if (blocks > 16384) blocks = 16384;
  ncr_edge_wmma_kernel<<<blocks, 256, 0, stream>>>(src, dst, Dmat, buf, accum, E);

  ncr_finalize_kernel<<<1, 32, 0, stream>>>(accum, (float*)d_out, E);
}